// GCNNet_72868415144434
// MI455X (gfx1250) — compile-verified
//
#include <hip/hip_runtime.h>
#include <hip/hip_bf16.h>

#define N_NODES  50000
#define N_EDGES  800000
#define N_GRAPHS 512

// padded feature widths (multiples of 16)
#define F0P 80     // 78  -> 80  (input / layer-1)
#define F1P 80     // 78  -> 80
#define F2P 160    // 156 -> 160
#define F3P 320    // 312 -> 320

typedef __attribute__((ext_vector_type(2))) float v2f;
typedef __attribute__((ext_vector_type(8))) float v8f;

// ---------------------------------------------------------------------------
// fp32 WMMA GEMM on fully padded operands:
//   C[M,N] = act(A[M,K] @ Bt[N,K]^T (+ bias))
// M,N,K all multiples of 16. Bt is the TRANSPOSED weight (row n = column n of
// B), so both A and Bt are read as contiguous aligned b64 per lane per K-step.
// One 16x16 tile per wave32; no guards, no divergence.
// ---------------------------------------------------------------------------
template<bool RELU_IN, bool RELU_OUT, bool HAS_BIAS>
__global__ void wmma_gemm(const float* __restrict__ A,
                          const float* __restrict__ Bt,
                          const float* __restrict__ bias,
                          float* __restrict__ C,
                          int M, int N, int K)
{
    const int lane  = threadIdx.x & 31;
    const int wave  = threadIdx.x >> 5;
    const int tilesN = N >> 4;
    const int tilesM = M >> 4;
    const int tile = blockIdx.x * (blockDim.x >> 5) + wave;
    if (tile >= tilesM * tilesN) return;       // wave-uniform

    const int tm  = tile / tilesN;
    const int tn  = tile % tilesN;
    const int llo = lane & 15;                 // lane within half-wave
    const int hi  = lane >> 4;                 // half-wave select

    // A: row (tm*16+llo), K elements {k0+2hi, k0+2hi+1} per step
    // Bt: row (tn*16+llo) holds column n of B along contiguous K
    const float* __restrict__ arow = A  + (size_t)((tm << 4) + llo) * K + (hi << 1);
    const float* __restrict__ brow = Bt + (size_t)((tn << 4) + llo) * K + (hi << 1);

    v8f acc = {};
    #pragma unroll 4
    for (int k0 = 0; k0 < K; k0 += 4) {
        v2f a = *(const v2f*)(arow + k0);      // global_load_b64
        if (RELU_IN) { a.x = fmaxf(a.x, 0.0f); a.y = fmaxf(a.y, 0.0f); }
        v2f b = *(const v2f*)(brow + k0);      // global_load_b64
        acc = __builtin_amdgcn_wmma_f32_16x16x4_f32(
                  false, a, false, b, (short)0, acc, false, false);
    }

    const int col = (tn << 4) + llo;
    const float bv = HAS_BIAS ? bias[col] : 0.0f;
    float* __restrict__ crow = C + (size_t)((tm << 4) + (hi << 3)) * N + col;
    #pragma unroll
    for (int v = 0; v < 8; ++v) {              // C/D layout: VGPR v -> M=v (+8 for hi half)
        float val = acc[v] + bv;
        if (RELU_OUT) val = fmaxf(val, 0.0f);
        crow[(size_t)v * N] = val;
    }
}

// ---------------------------------------------------------------------------
// Weight pad+transpose: Wt[Npad x Kpad], Wt[n,k] = (n<N && k<K) ? W[k,n] : 0
// ---------------------------------------------------------------------------
__global__ void pad_transpose(const float* __restrict__ W, float* __restrict__ Wt,
                              int K, int N, int Kpad, int Npad)
{
    int idx = blockIdx.x * blockDim.x + threadIdx.x;
    if (idx >= Kpad * Npad) return;
    int n = idx / Kpad;
    int k = idx % Kpad;
    Wt[idx] = (n < N && k < K) ? W[(size_t)k * N + n] : 0.0f;
}

// Row-wise zero-pad: Xp[r,c] = c<C ? X[r,c] : 0   (also used for bias, rows=1)
__global__ void pad_rows(const float* __restrict__ X, float* __restrict__ Xp,
                         int rows, int C, int Cpad)
{
    size_t idx = (size_t)blockIdx.x * blockDim.x + threadIdx.x;
    if (idx >= (size_t)rows * Cpad) return;
    int r = (int)(idx / Cpad);
    int c = (int)(idx % Cpad);
    Xp[idx] = (c < C) ? X[(size_t)r * C + c] : 0.0f;
}

// ---------------------------------------------------------------------------
// Degree / normalization (deg includes self-loop; in-place -> deg^-1/2)
// ---------------------------------------------------------------------------
__global__ void deg_init(float* __restrict__ deg)
{
    int i = blockIdx.x * blockDim.x + threadIdx.x;
    if (i < N_NODES) deg[i] = 1.0f;
}
__global__ void deg_count(const int* __restrict__ dst, float* __restrict__ deg)
{
    int e = blockIdx.x * blockDim.x + threadIdx.x;
    if (e < N_EDGES) atomicAdd(&deg[dst[e]], 1.0f);
}
__global__ void deg_rsqrt(float* __restrict__ deg)
{
    int i = blockIdx.x * blockDim.x + threadIdx.x;
    if (i < N_NODES) deg[i] = rsqrtf(deg[i]);
}

// ---------------------------------------------------------------------------
// O[i,f] = H[i,f] * dinv[i]^2 + b[f]  (self-loop term + bias; F padded)
// ---------------------------------------------------------------------------
__global__ void self_bias(const float* __restrict__ H,
                          const float* __restrict__ dinv,
                          const float* __restrict__ b,
                          float* __restrict__ O, int F)
{
    size_t i = (size_t)blockIdx.x * blockDim.x + threadIdx.x;
    if (i >= (size_t)N_NODES * F) return;
    int node = (int)(i / F);
    float dv = dinv[node];
    O[i] = H[i] * dv * dv + b[(int)(i % F)];
}

// ---------------------------------------------------------------------------
// Edge aggregation: O[dst] += H[src] * dinv[src]*dinv[dst]; one wave per edge.
// Atomic adds stay in L2 (output buffer << 192 MB L2). Prefetch ahead.
// ---------------------------------------------------------------------------
__global__ void edge_agg(const int* __restrict__ src,
                         const int* __restrict__ dst,
                         const float* __restrict__ H,
                         const float* __restrict__ dinv,
                         float* __restrict__ O, int F)
{
    int e    = (blockIdx.x * blockDim.x + threadIdx.x) >> 5;
    int lane = threadIdx.x & 31;
    if (e >= N_EDGES) return;

    int ep = e + 64;                                   // global_prefetch_b8
    if (ep < N_EDGES && lane == 0)
        __builtin_prefetch(&H[(size_t)src[ep] * F], 0, 0);

    const int s = src[e];
    const int d = dst[e];
    const float w = dinv[s] * dinv[d];
    const float* __restrict__ hrow = H + (size_t)s * F;
    float* __restrict__ orow = O + (size_t)d * F;
    for (int f = lane; f < F; f += 32)
        atomicAdd(&orow[f], hrow[f] * w);
}

// ---------------------------------------------------------------------------
// Per-graph max pool (sorted even batch split); applies layer-3 ReLU.
// Pad columns of the input are all zero -> pad columns of G become 0.
// ---------------------------------------------------------------------------
__global__ void pool_max(const float* __restrict__ O3, float* __restrict__ G)
{
    const int g = blockIdx.x;
    const int start = (g * N_NODES + (N_GRAPHS - 1)) / N_GRAPHS;
    const int end   = ((g + 1) * N_NODES + (N_GRAPHS - 1)) / N_GRAPHS;
    for (int f = threadIdx.x; f < F3P; f += blockDim.x) {
        float m = -INFINITY;
        for (int i = start; i < end; ++i)
            m = fmaxf(m, O3[(size_t)i * F3P + f]);
        G[(size_t)g * F3P + f] = fmaxf(m, 0.0f);       // relu(max) == max(relu)
    }
}

// ---------------------------------------------------------------------------
template<bool RI, bool RO, bool HB>
static inline void launch_gemm(const float* A, const float* Bt, const float* bias,
                               float* C, int M, int N, int K, hipStream_t stream)
{
    const int tiles  = (M >> 4) * (N >> 4);
    const int blocks = (tiles + 7) / 8;                // 8 waves / 256 threads per block
    wmma_gemm<RI, RO, HB><<<blocks, 256, 0, stream>>>(A, Bt, bias, C, M, N, K);
}

extern "C" void kernel_launch(void* const* d_in, const int* in_sizes, int n_in,
                              void* d_out, int out_size, void* d_ws, size_t ws_size,
                              hipStream_t stream) {
    (void)in_sizes; (void)n_in; (void)out_size; (void)ws_size;

    const float* x   = (const float*)d_in[0];          // [50000, 78]
    const int*   ei  = (const int*)  d_in[1];          // [2, 800000]
    const float* W1  = (const float*)d_in[3];   const float* b1  = (const float*)d_in[4];
    const float* W2  = (const float*)d_in[5];   const float* b2  = (const float*)d_in[6];
    const float* W3  = (const float*)d_in[7];   const float* b3  = (const float*)d_in[8];
    const float* Wf1 = (const float*)d_in[9];   const float* bf1 = (const float*)d_in[10];
    const float* Wf2 = (const float*)d_in[11];  const float* bf2 = (const float*)d_in[12];
    float* out = (float*)d_out;                        // [512, 128]

    const int* src = ei;
    const int* dst = ei + N_EDGES;

    // ---- workspace layout (floats; every region a multiple of 16 elems) ----
    float* p = (float*)d_ws;
    float* xp   = p; p += (size_t)N_NODES * F0P;       // padded input [50000,80]
    float* bufP = p; p += (size_t)N_NODES * F3P;       // ping [50000,320]
    float* bufQ = p; p += (size_t)N_NODES * F3P;       // pong [50000,320]
    float* dinv = p; p += N_NODES;
    float* G    = p; p += (size_t)N_GRAPHS * F3P;      // pooled [512,320]
    float* G2   = p; p += (size_t)N_GRAPHS * 1024;     // fc1 out [512,1024]
    float* W1t  = p; p += F1P * F0P;                   // [80,80]
    float* W2t  = p; p += F2P * F1P;                   // [160,80]
    float* W3t  = p; p += F3P * F2P;                   // [320,160]
    float* Wf1t = p; p += 1024 * F3P;                  // [1024,320]
    float* Wf2t = p; p += 128 * 1024;                  // [128,1024]
    float* b1p  = p; p += F1P;
    float* b2p  = p; p += F2P;
    float* b3p  = p; p += F3P;

    const int T = 256;
    #define GRID1D(n) ((unsigned)(((size_t)(n) + T - 1) / T))

    // ---- one-time (per launch) pad / transpose of operands ----
    pad_rows     <<<GRID1D((size_t)N_NODES * F0P), T, 0, stream>>>(x,  xp,  N_NODES, 78, F0P);
    pad_transpose<<<GRID1D(F1P * F0P), T, 0, stream>>>(W1,  W1t,  78,  78,   F0P, F1P);
    pad_transpose<<<GRID1D(F2P * F1P), T, 0, stream>>>(W2,  W2t,  78,  156,  F1P, F2P);
    pad_transpose<<<GRID1D(F3P * F2P), T, 0, stream>>>(W3,  W3t,  156, 312,  F2P, F3P);
    pad_transpose<<<GRID1D(1024 * F3P), T, 0, stream>>>(Wf1, Wf1t, 312, 1024, F3P, 1024);
    pad_transpose<<<GRID1D(128 * 1024), T, 0, stream>>>(Wf2, Wf2t, 1024, 128, 1024, 128);
    pad_rows     <<<GRID1D(F1P), T, 0, stream>>>(b1, b1p, 1, 78,  F1P);
    pad_rows     <<<GRID1D(F2P), T, 0, stream>>>(b2, b2p, 1, 156, F2P);
    pad_rows     <<<GRID1D(F3P), T, 0, stream>>>(b3, b3p, 1, 312, F3P);

    // ---- degree / normalization ----
    deg_init <<<GRID1D(N_NODES), T, 0, stream>>>(dinv);
    deg_count<<<GRID1D(N_EDGES), T, 0, stream>>>(dst, dinv);
    deg_rsqrt<<<GRID1D(N_NODES), T, 0, stream>>>(dinv);

    const unsigned aggBlocks = GRID1D((size_t)N_EDGES * 32);

    // ---- layer 1: 80 -> 80 ----
    launch_gemm<false, false, false>(xp, W1t, nullptr, bufQ, N_NODES, F1P, F0P, stream);
    self_bias<<<GRID1D((size_t)N_NODES * F1P), T, 0, stream>>>(bufQ, dinv, b1p, bufP, F1P);
    edge_agg <<<aggBlocks, T, 0, stream>>>(src, dst, bufQ, dinv, bufP, F1P);

    // ---- layer 2: 80 -> 160 (prev ReLU fused into A-load) ----
    launch_gemm<true, false, false>(bufP, W2t, nullptr, bufQ, N_NODES, F2P, F1P, stream);
    self_bias<<<GRID1D((size_t)N_NODES * F2P), T, 0, stream>>>(bufQ, dinv, b2p, bufP, F2P);
    edge_agg <<<aggBlocks, T, 0, stream>>>(src, dst, bufQ, dinv, bufP, F2P);

    // ---- layer 3: 160 -> 320 ----
    launch_gemm<true, false, false>(bufP, W3t, nullptr, bufQ, N_NODES, F3P, F2P, stream);
    self_bias<<<GRID1D((size_t)N_NODES * F3P), T, 0, stream>>>(bufQ, dinv, b3p, bufP, F3P);
    edge_agg <<<aggBlocks, T, 0, stream>>>(src, dst, bufQ, dinv, bufP, F3P);

    // ---- global max pool (applies layer-3 ReLU) -> [512, 320] ----
    pool_max<<<N_GRAPHS, T, 0, stream>>>(bufP, G);

    // ---- FC head ----
    launch_gemm<false, true,  true>(G,  Wf1t, bf1, G2,  N_GRAPHS, 1024, F3P,  stream);
    launch_gemm<false, false, true>(G2, Wf2t, bf2, out, N_GRAPHS, 128,  1024, stream);

    #undef GRID1D
}